// Proposal_Layer_60404420051297
// MI455X (gfx1250) — compile-verified
//
#include <hip/hip_runtime.h>
#include <hip/hip_bf16.h>
#include <stdint.h>

// ---------------- problem constants ----------------
#define N_ANCH   140625      // 125*125*9
#define KSEL     6000        // PRE_NMS_TOPK
#define OUTK     300         // AFT_NMS_TOPK
#define NMS_THR  0.7f
#define NBLK     94          // ceil(KSEL/64)
#define SORT_N   8192        // candidate sort capacity (power of 2 >= KSEL)

// ---------------- CDNA5 async global->LDS helpers ----------------
#if __has_builtin(__builtin_amdgcn_global_load_async_to_lds_b128)
#define HAVE_ASYNC_LDS 1
#else
#define HAVE_ASYNC_LDS 0
#endif

typedef int v4i __attribute__((vector_size(16)));

#define AS_GLOBAL __attribute__((address_space(1)))
#define AS_LDS    __attribute__((address_space(3)))

__device__ __forceinline__ void async_ld_b128(void* lds, const void* g) {
#if HAVE_ASYNC_LDS
  __builtin_amdgcn_global_load_async_to_lds_b128(
      (AS_GLOBAL v4i*)(g), (AS_LDS v4i*)(lds), 0, 0);
#else
  *(float4*)lds = *(const float4*)g;
#endif
}

__device__ __forceinline__ void async_ld_b32(void* lds, const void* g) {
#if HAVE_ASYNC_LDS && __has_builtin(__builtin_amdgcn_global_load_async_to_lds_b32)
  __builtin_amdgcn_global_load_async_to_lds_b32(
      (AS_GLOBAL int*)(g), (AS_LDS int*)(lds), 0, 0);
#else
  *(float*)lds = *(const float*)g;
#endif
}

__device__ __forceinline__ void wait_async_lds() {
#if __has_builtin(__builtin_amdgcn_s_wait_asynccnt)
  __builtin_amdgcn_s_wait_asynccnt(0);
#else
  asm volatile("s_wait_asynccnt 0" ::: "memory");
#endif
}

// ---------------- kernels ----------------

// state[0]=prefix, state[1]=prefix mask, state[2]=k still needed; also zero hist
__global__ void k_init(unsigned* state, unsigned* candcnt, unsigned* hist) {
  hist[threadIdx.x] = 0u;
  if (threadIdx.x == 0) {
    state[0] = 0u; state[1] = 0u; state[2] = (unsigned)KSEL;
    candcnt[0] = 0u;
  }
}

// decode + clip boxes, build monotonic sortable keys from fg scores
__global__ void k_decode(const float* __restrict__ score,
                         const float* __restrict__ reg,
                         const float* __restrict__ anc,
                         const float* __restrict__ im_info,
                         float4* __restrict__ boxes,
                         unsigned* __restrict__ keys, int n) {
  int i = blockIdx.x * blockDim.x + threadIdx.x;
  if (i >= n) return;
  float4 a = ((const float4*)anc)[i];
  float4 d = ((const float4*)reg)[i];
  float w  = a.z - a.x + 1.0f;
  float h  = a.w - a.y + 1.0f;
  float cx = a.x + 0.5f * w;
  float cy = a.y + 0.5f * h;
  float pcx = d.x * w + cx;
  float pcy = d.y * h + cy;
  float pw  = w * expf(d.z);
  float ph  = h * expf(d.w);
  float x1 = pcx - 0.5f * pw, y1 = pcy - 0.5f * ph;
  float x2 = pcx + 0.5f * pw, y2 = pcy + 0.5f * ph;
  float xm = im_info[0] - 1.0f;   // reference clips x with im_info[0]
  float ym = im_info[1] - 1.0f;   // and y with im_info[1]
  x1 = fminf(fmaxf(x1, 0.0f), xm);
  x2 = fminf(fmaxf(x2, 0.0f), xm);
  y1 = fminf(fmaxf(y1, 0.0f), ym);
  y2 = fminf(fmaxf(y2, 0.0f), ym);
  boxes[i] = make_float4(x1, y1, x2, y2);
  float s = ((const float2*)score)[i].x;       // score[0, i, 0], b64 load
  unsigned k = __float_as_uint(s);
  k = (k & 0x80000000u) ? ~k : (k | 0x80000000u);  // order-preserving map
  keys[i] = k;
}

// per-block LDS histogram -> few global atomics
__global__ __launch_bounds__(256)
void k_hist(const unsigned* __restrict__ keys, int n,
            const unsigned* __restrict__ state,
            unsigned* __restrict__ hist, int shift) {
  __shared__ unsigned lh[256];
  lh[threadIdx.x] = 0u;
  __syncthreads();
  unsigned prefix = state[0], pmask = state[1];
  for (int i = blockIdx.x * blockDim.x + threadIdx.x; i < n;
       i += gridDim.x * blockDim.x) {
    unsigned k = keys[i];
    if ((k & pmask) == prefix)
      atomicAdd(&lh[(k >> shift) & 255u], 1u);
  }
  __syncthreads();
  unsigned v = lh[threadIdx.x];
  if (v) atomicAdd(&hist[threadIdx.x], v);
}

// pick highest byte bucket so count(key > threshold) stays < k_needed;
// also re-zero hist for the next pass.
__global__ __launch_bounds__(256)
void k_scan(unsigned* hist, unsigned* state, int shift) {
  __shared__ unsigned h[256];
  int t = threadIdx.x;
  h[t] = hist[t];
  hist[t] = 0u;                      // ready for next pass
  __syncthreads();
  if (t == 0) {
    unsigned kneed = state[2];
    unsigned cum = 0;
    int b = 255;
    for (; b > 0; --b) {
      unsigned c = h[b];
      if (cum + c >= kneed) break;
      cum += c;
    }
    state[0] |= ((unsigned)b) << shift;
    state[1] |= 0xFFu << shift;
    state[2] = kneed - cum;
  }
}

__global__ void k_compact(const unsigned* __restrict__ keys, int n,
                          const unsigned* __restrict__ state,
                          unsigned* __restrict__ candkey, int* __restrict__ candidx,
                          unsigned* __restrict__ candcnt) {
  unsigned T = state[0];  // exact 32-bit threshold after 4 passes
  for (int i = blockIdx.x * blockDim.x + threadIdx.x; i < n;
       i += gridDim.x * blockDim.x) {
    unsigned k = keys[i];
    if (k >= T) {
      unsigned pos = atomicAdd(candcnt, 1u);
      if (pos < SORT_N) { candkey[pos] = k; candidx[pos] = i; }
    }
  }
}

// single-workgroup bitonic sort of <=8192 (key,idx) pairs in 64KB LDS.
// composite = (key<<32) | ~idx -> descending sort == (value desc, index asc),
// exactly matching lax.top_k tie-breaking.
__global__ __launch_bounds__(1024)
void k_sort(const unsigned* __restrict__ candkey, const int* __restrict__ candidx,
            const unsigned* __restrict__ candcnt, int* __restrict__ order) {
  __shared__ unsigned long long sk[SORT_N];   // 64 KB (of 320 KB/WGP on CDNA5)
  int tid = threadIdx.x, nt = blockDim.x;
  unsigned cnt = candcnt[0];
  if (cnt > SORT_N) cnt = SORT_N;
  for (int i = tid; i < SORT_N; i += nt) {
    sk[i] = (i < (int)cnt)
        ? ((((unsigned long long)candkey[i]) << 32) |
           (unsigned)(~(unsigned)candidx[i]))
        : 0ull;  // padding sorts last
  }
  __syncthreads();
  for (int k = 2; k <= SORT_N; k <<= 1) {
    for (int j = k >> 1; j > 0; j >>= 1) {
      for (int i = tid; i < SORT_N; i += nt) {
        int ixj = i ^ j;
        if (ixj > i) {
          unsigned long long a = sk[i], b = sk[ixj];
          bool up = ((i & k) == 0);              // descending overall
          bool sw = up ? (a < b) : (a > b);
          if (sw) { sk[i] = b; sk[ixj] = a; }
        }
      }
      __syncthreads();
    }
  }
  for (int i = tid; i < KSEL; i += nt)
    order[i] = (int)(~(unsigned)(sk[i] & 0xFFFFFFFFull));
}

__global__ void k_gather(const float4* __restrict__ boxes,
                         const int* __restrict__ order,
                         float4* __restrict__ sbox, float* __restrict__ sarea) {
  int i = blockIdx.x * blockDim.x + threadIdx.x;
  if (i >= KSEL) return;
  float4 b = boxes[order[i]];
  sbox[i] = b;
  sarea[i] = (b.z - b.x + 1.0f) * (b.w - b.y + 1.0f);
}

// 94x94 tile grid; each 64-thread block computes one 64x64 tile of the
// suppression bitmask. Column boxes staged to LDS via CDNA5 async loads.
__global__ __launch_bounds__(64)
void k_nms_mask(const float4* __restrict__ sbox, const float* __restrict__ sarea,
                unsigned long long* __restrict__ mask) {
  int t  = threadIdx.x;
  int rg = blockIdx.y * 64 + t;
  // tiles fully below the diagonal can never set bits (need c > r): store 0
  if (blockIdx.x < blockIdx.y) {
    if (rg < KSEL) mask[(size_t)rg * NBLK + blockIdx.x] = 0ull;
    return;
  }
  __shared__ float4 cbox[64];
  __shared__ float  carea[64];
  int c0 = blockIdx.x * 64;
  int cg = c0 + t;
  if (cg < KSEL) {
    async_ld_b128(&cbox[t], &sbox[cg]);   // ASYNCcnt path (gfx1250)
    async_ld_b32(&carea[t], &sarea[cg]);
  } else {
    cbox[t]  = make_float4(0.f, 0.f, 0.f, 0.f);
    carea[t] = 0.f;
  }
  wait_async_lds();                        // s_wait_asynccnt 0
  __syncthreads();

  if (rg >= KSEL) return;
  float4 rb = sbox[rg];
  float  ra = sarea[rg];
  unsigned long long bits = 0ull;
#pragma unroll 4
  for (int c = 0; c < 64; ++c) {
    int cg2 = c0 + c;
    if (cg2 > rg && cg2 < KSEL) {
      float4 cb = cbox[c];
      float iw = fmaxf(fminf(rb.z, cb.z) - fmaxf(rb.x, cb.x) + 1.0f, 0.0f);
      float ih = fmaxf(fminf(rb.w, cb.w) - fmaxf(rb.y, cb.y) + 1.0f, 0.0f);
      float inter = iw * ih;
      float iou = inter / (ra + carea[c] - inter);
      if (iou > NMS_THR) bits |= (1ull << c);
    }
  }
  mask[(size_t)rg * NBLK + blockIdx.x] = bits;
}

// single-wave32 sequential greedy reduce: 94 suppression words live in 3
// registers per lane; current word broadcast via __shfl (no barriers on the
// 6000-iteration serial chain). Exact reference greedy semantics.
__global__ __launch_bounds__(32)
void k_nms_reduce_out(const unsigned long long* __restrict__ mask,
                      const float4* __restrict__ sbox,
                      float* __restrict__ out) {
  __shared__ int sel[OUTK];
  int l = threadIdx.x;                 // 0..31, one wave
  unsigned long long r0 = 0ull, r1 = 0ull, r2 = 0ull;  // words l, 32+l, 64+l
  int cnt = 0;                         // uniform across lanes
  for (int i = 0; i < KSEL; ++i) {
    // prefetch next row onto the serial chain (global_prefetch_b8)
    if (i + 1 < KSEL) {
      const char* nxt = (const char*)(mask + (size_t)(i + 1) * NBLK);
      __builtin_prefetch(nxt + 8 * l, 0, 3);
      __builtin_prefetch(nxt + 8 * (32 + l), 0, 3);
      if (l < NBLK - 64) __builtin_prefetch(nxt + 8 * (64 + l), 0, 3);
    }
    int w = i >> 6, bit = i & 63;
    unsigned long long rw = (w < 32) ? r0 : (w < 64) ? r1 : r2;
    unsigned long long word = __shfl(rw, w & 31, 32);
    if (!((word >> bit) & 1ull)) {     // box i survives
      const unsigned long long* row = mask + (size_t)i * NBLK;
      r0 |= row[l];
      r1 |= row[32 + l];
      if (l < NBLK - 64) r2 |= row[64 + l];
      if (l == 0 && cnt < OUTK) sel[cnt] = i;
      cnt++;
    }
  }
  __syncthreads();
  int n = cnt;
  for (int idx = l; idx < OUTK * 4; idx += 32) {
    int j = idx >> 2, k = idx & 3;
    float v = 0.0f;
    if (j < n) {
      float4 b = sbox[sel[j]];
      v = (k == 0) ? b.x : (k == 1) ? b.y : (k == 2) ? b.z : b.w;
    }
    out[idx] = v;
  }
}

// ---------------- host-side launch sequence ----------------
extern "C" void kernel_launch(void* const* d_in, const int* in_sizes, int n_in,
                              void* d_out, int out_size, void* d_ws, size_t ws_size,
                              hipStream_t stream) {
  const float* score   = (const float*)d_in[0];   // (1, N, 2)
  const float* reg     = (const float*)d_in[1];   // (N, 4)
  const float* anchors = (const float*)d_in[2];   // (N, 4)
  const float* im_info = (const float*)d_in[3];   // (2,)
  float* out = (float*)d_out;                     // (300, 4)
  const int N = in_sizes[2] / 4;                  // == N_ANCH

  char* ws = (char*)d_ws;
  size_t off = 0;
  auto take = [&](size_t bytes) -> char* {
    char* p = ws + off;
    off = (off + bytes + 255) & ~(size_t)255;
    return p;
  };
  float4*   boxes   = (float4*)  take((size_t)N * 16);
  unsigned* keys    = (unsigned*)take((size_t)N * 4);
  unsigned* state   = (unsigned*)take(16);
  unsigned* hist    = (unsigned*)take(256 * 4);
  unsigned* candcnt = (unsigned*)take(16);
  unsigned* candkey = (unsigned*)take((size_t)SORT_N * 4);
  int*      candidx = (int*)     take((size_t)SORT_N * 4);
  int*      order   = (int*)     take((size_t)KSEL * 4);
  float4*   sbox    = (float4*)  take((size_t)KSEL * 16);
  float*    sarea   = (float*)   take((size_t)KSEL * 4);
  unsigned long long* mask =
      (unsigned long long*)take((size_t)KSEL * NBLK * 8);
  (void)ws_size; (void)n_in; (void)out_size;

  k_init<<<1, 256, 0, stream>>>(state, candcnt, hist);
  k_decode<<<(N + 255) / 256, 256, 0, stream>>>(score, reg, anchors, im_info,
                                                boxes, keys, N);
  for (int p = 0; p < 4; ++p) {
    int shift = 24 - 8 * p;
    k_hist<<<128, 256, 0, stream>>>(keys, N, state, hist, shift);
    k_scan<<<1, 256, 0, stream>>>(hist, state, shift);
  }
  k_compact<<<256, 256, 0, stream>>>(keys, N, state, candkey, candidx, candcnt);
  k_sort<<<1, 1024, 0, stream>>>(candkey, candidx, candcnt, order);
  k_gather<<<(KSEL + 255) / 256, 256, 0, stream>>>(boxes, order, sbox, sarea);
  k_nms_mask<<<dim3(NBLK, NBLK), 64, 0, stream>>>(sbox, sarea, mask);
  k_nms_reduce_out<<<1, 32, 0, stream>>>(mask, sbox, out);
}